// CompressSFA_77395310674146
// MI455X (gfx1250) — compile-verified
//
#include <hip/hip_runtime.h>
#include <cstddef>

// CDNA5 / gfx1250, wave32. fp32 WMMA: V_WMMA_F32_16X16X4_F32
typedef __attribute__((ext_vector_type(2))) float v2f;
typedef __attribute__((ext_vector_type(8))) float v8f;

// Explicit global (addrspace(1)) pointer so gathered loads lower to
// global_load_* (GVS: SGPR base + VGPR offset) instead of flat_load_*.
typedef const float __attribute__((address_space(1)))* gfp;

#define NEG_BIG (-1e30f)

__device__ __forceinline__ float xor16(float x) {
    return __shfl_xor(x, 16, 32);
}

__device__ __forceinline__ gfp uniform_base(const float* p) {
    unsigned long long u = (unsigned long long)p;
    unsigned lo = (unsigned)__builtin_amdgcn_readfirstlane((int)(unsigned)u);
    unsigned hi = (unsigned)__builtin_amdgcn_readfirstlane((int)(unsigned)(u >> 32));
    return (gfp)(((unsigned long long)hi << 32) | lo);
}

// Phase 1: one block per (b*Hkv, L-split). 8 waves split the 16-key tiles.
// Each wave: S^T = K_tile x Q^T via wmma_f32_16x16x4_f32, online softmax
// per lane-column (query group g), O^T += V^T x P^T via the same WMMA.
// Waves merged through LDS; block writes an unnormalized flash partial
// (O 1024 floats, m[8], l[8]) to ws.
__global__ __launch_bounds__(256) void sfa_phase1(
    const float* __restrict__ query,
    const float* __restrict__ ori_kv,
    const float* __restrict__ cmp_kv,
    const int* __restrict__ ori_bt,
    const int* __restrict__ cmp_bt,
    const int* __restrict__ sink_arr,
    const int* __restrict__ seq_arr,
    const int* __restrict__ cmp_idx,
    const float* __restrict__ scale_p,
    const int* __restrict__ win_p,
    const int* __restrict__ ratio_p,
    float* __restrict__ ws,
    int H, int Hkv, int maxblk, int maxcblk, int nsel, int nsplit)
{
    const int bh    = blockIdx.x;     // b*Hkv + hkv
    const int split = blockIdx.y;
    const int b     = bh / Hkv;
    const int hkv   = bh % Hkv;
    const int tid   = threadIdx.x;
    const int wave  = tid >> 5;
    const int lane  = tid & 31;
    const int lhalf = lane & 15;
    const bool hi   = lane >= 16;
    const int Greal = H / Hkv;        // 8

    const float scale = scale_p[0];
    const int win   = win_p[0];
    const int ratio = ratio_p[0];
    const int seq   = seq_arr[b];
    const int sink  = sink_arr[b];
    const int L      = 128 + win + nsel;     // 896
    const int ntiles = L >> 4;               // 56
    const int tps    = (ntiles + nsplit - 1) / nsplit;
    const int t0     = split * tps;
    const int t1     = min(ntiles, t0 + tps);

    // ---- Load Q^T B-operand chunks: qv[c] pairs with d = 4c..4c+3 ----
    // B 4x16 layout: VGPR0 = {K=0 row | K=2 row}, VGPR1 = {K=1 | K=3}
    // lane n (lo) -> Q[g=n][4c],Q[g=n][4c+1]; lane n (hi) -> Q[g][4c+2],[4c+3]
    v2f qv[32];
    {
        gfp qb = uniform_base(query);
        const int qoff = ((b * H + hkv * Greal + lhalf) << 7) + (hi ? 2 : 0);
#pragma unroll
        for (int c = 0; c < 32; ++c) {
            v2f q; q.x = 0.f; q.y = 0.f;
            if (lhalf < Greal) {
                gfp p = qb + qoff + 4 * c;
                q.x = p[0]; q.y = p[1];
            }
            qv[c] = q;
        }
    }

    v8f o[8];
#pragma unroll
    for (int i = 0; i < 8; ++i) o[i] = (v8f){0.f,0.f,0.f,0.f,0.f,0.f,0.f,0.f};
    float m_run = NEG_BIG, l_run = 0.f;

    const int rowstride = 2 * Hkv * 128;   // elems per token (1024)

    for (int t = t0 + wave; t < t1; t += 8) {
        // ---- lanes 0..15 (mirrored in 16..31) compute key row offset ----
        // Regions are 16-aligned, so `cache` is wave-uniform per tile.
        const int idx = (t << 4) + lhalf;
        const float* cache;
        const int* table;
        int pos, valid;
        if (idx < 128) {                       // attention sink region
            pos = idx;
            valid = (pos < sink) && (pos < seq);
            cache = ori_kv; table = ori_bt + (size_t)b * maxblk;
        } else if (idx < 128 + win) {          // sliding window region
            pos = seq - win + (idx - 128);
            valid = (pos >= 0) && (pos >= sink);
            cache = ori_kv; table = ori_bt + (size_t)b * maxblk;
        } else {                               // selected compressed region
            int c = cmp_idx[(size_t)b * nsel + (idx - 128 - win)];
            valid = c < (seq / ratio);
            pos = c;
            cache = cmp_kv; table = cmp_bt + (size_t)b * maxcblk;
        }
        pos = max(pos, 0);
        const int blk = table[pos >> 7];       // block_size = 128
        // 32-bit element offset of this key's K row inside its cache
        const int rowoff = (blk * 128 + (pos & 127)) * rowstride + hkv * 128;

        // wave-uniform 16-bit valid mask
#if __has_builtin(__builtin_amdgcn_ballot_w32)
        const unsigned mask16 =
            ((unsigned)__builtin_amdgcn_ballot_w32(valid != 0)) & 0xFFFFu;
#else
        unsigned mask16 = 0;
        {
            int vb = valid ? 1 : 0;
#pragma unroll
            for (int i = 0; i < 16; ++i)
                mask16 |= ((unsigned)__shfl(vb, i, 32) & 1u) << i;
        }
#endif
        if (mask16 == 0) continue;             // uniform: EXEC stays full

        gfp kb = uniform_base(cache);          // SGPR base -> GVS global loads
        const int koff = __shfl(rowoff, lhalf, 32);
        // Prefetch this key's K row and V row (each 512 B) into cache.
        __builtin_prefetch((const void*)(kb + rowoff), 0, 0);
        __builtin_prefetch((const void*)(kb + rowoff + Hkv * 128), 0, 0);

        // ---- S^T = K_tile(16x128) x Q^T(128x16), 32 chained K=4 WMMAs ----
        // A 16x4 layout: lane m -> row m; VGPR0 = K={0|2}, VGPR1 = K={1|3}
        const int dof = hi ? 2 : 0;
        v2f av[32];
#pragma unroll
        for (int c = 0; c < 32; ++c) {
            gfp p = kb + koff + 4 * c + dof;
            av[c].x = p[0];
            av[c].y = p[1];
        }
        v8f s = (v8f){0.f,0.f,0.f,0.f,0.f,0.f,0.f,0.f};
#pragma unroll
        for (int c = 0; c < 32; ++c) {
            s = __builtin_amdgcn_wmma_f32_16x16x4_f32(
                    false, av[c], false, qv[c], (short)0, s, false, false);
        }

        // ---- online softmax over key rows (per lane-column g) ----
        float mt = NEG_BIG;
        v8f sv;
#pragma unroll
        for (int j = 0; j < 8; ++j) {
            bool ok = (mask16 >> (j + (hi ? 8 : 0))) & 1u;
            float x = ok ? s[j] * scale : NEG_BIG;
            sv[j] = x;
            mt = fmaxf(mt, x);
        }
        mt = fmaxf(mt, xor16(mt));
        const float m_new = fmaxf(m_run, mt);
        const float f = __expf(m_run - m_new);
        float lsum = 0.f;
        v8f pj;
#pragma unroll
        for (int j = 0; j < 8; ++j) {
            float e = __expf(sv[j] - m_new);
            pj[j] = e;
            lsum += e;
        }
        lsum += xor16(lsum);
        l_run = l_run * f + lsum;
        m_run = m_new;
#pragma unroll
        for (int nt = 0; nt < 8; ++nt)
#pragma unroll
            for (int j = 0; j < 8; ++j) o[nt][j] *= f;

        // ---- O^T += V^T(16d x 4k) x P^T(4k x 16g) ----
#pragma unroll
        for (int kc = 0; kc < 4; ++kc) {
            // Build P^T B-operand from S^T C/D layout (half-lane swaps)
            v2f bm;
            if (kc < 2) {
                float s0 = xor16(pj[4 * kc + 2]);
                float s1 = xor16(pj[4 * kc + 3]);
                bm.x = hi ? s0 : pj[4 * kc];
                bm.y = hi ? s1 : pj[4 * kc + 1];
            } else {
                const int r = 4 * kc - 8;
                float s0 = xor16(pj[r]);
                float s1 = xor16(pj[r + 1]);
                bm.x = hi ? pj[r + 2] : s0;
                bm.y = hi ? pj[r + 3] : s1;
            }
            const int k0 = 4 * kc + (hi ? 2 : 0);
            const int voff0 = __shfl(rowoff, k0, 32)     + Hkv * 128;
            const int voff1 = __shfl(rowoff, k0 + 1, 32) + Hkv * 128;
            // Preload the 8 A-operands of this k-chunk, then 8 independent
            // WMMAs into the 8 d-tile accumulators.
            v2f avv[8];
#pragma unroll
            for (int nt = 0; nt < 8; ++nt) {
                avv[nt].x = kb[voff0 + nt * 16 + lhalf];
                avv[nt].y = kb[voff1 + nt * 16 + lhalf];
            }
#pragma unroll
            for (int nt = 0; nt < 8; ++nt) {
                o[nt] = __builtin_amdgcn_wmma_f32_16x16x4_f32(
                            false, avv[nt], false, bm, (short)0, o[nt],
                            false, false);
            }
        }
    }

    // ---- merge 8 waves via LDS (store only the 8 real g columns) ----
    __shared__ float lds_o[8][1024];  // [wave][nt*128 + j*16 + sub]
    __shared__ float lds_m[8][8];
    __shared__ float lds_l[8][8];
    if ((lane & 15) < 8) {
        const int sub = ((lane >> 4) << 3) + (lane & 7);
#pragma unroll
        for (int nt = 0; nt < 8; ++nt)
#pragma unroll
            for (int j = 0; j < 8; ++j)
                lds_o[wave][nt * 128 + j * 16 + sub] = o[nt][j];
        if (!hi) { lds_m[wave][lane] = m_run; lds_l[wave][lane] = l_run; }
    }
    __syncthreads();

    float* chunk = ws + ((size_t)bh * nsplit + split) * 1040;
    const int GrD = Greal * 128;  // 1024
    for (int e = tid; e < GrD; e += 256) {
        const int g = e >> 7, d = e & 127;
        float M = NEG_BIG;
#pragma unroll
        for (int w = 0; w < 8; ++w) M = fmaxf(M, lds_m[w][g]);
        const int nt = d >> 4, j = d & 15;
        const int sub = (j < 8) ? g : (g + 8);
        const int jj = j & 7;
        float num = 0.f, den = 0.f;
#pragma unroll
        for (int w = 0; w < 8; ++w) {
            float aw = __expf(lds_m[w][g] - M);
            den += aw * lds_l[w][g];
            num += aw * lds_o[w][nt * 128 + jj * 16 + sub];
        }
        chunk[e] = num;
        if (d == 0) { chunk[1024 + g] = M; chunk[1032 + g] = den; }
    }
}

// Phase 2: merge the L-splits, normalize, write out[b, h, d].
__global__ __launch_bounds__(256) void sfa_phase2(
    const float* __restrict__ ws, float* __restrict__ out,
    int H, int Hkv, int nsplit)
{
    const int bh = blockIdx.x;
    const int b = bh / Hkv, hkv = bh % Hkv;
    const int Greal = H / Hkv;
    const int GrD = Greal * 128;
    const float* c0 = ws + (size_t)bh * nsplit * 1040;
    for (int e = threadIdx.x; e < GrD; e += 256) {
        const int g = e >> 7, d = e & 127;
        float M = NEG_BIG;
        for (int s = 0; s < nsplit; ++s)
            M = fmaxf(M, c0[s * 1040 + 1024 + g]);
        float num = 0.f, den = 0.f;
        for (int s = 0; s < nsplit; ++s) {
            float aw = __expf(c0[s * 1040 + 1024 + g] - M);
            num += aw * c0[s * 1040 + e];
            den += aw * c0[s * 1040 + 1032 + g];
        }
        out[((size_t)b * H + (size_t)hkv * Greal + g) * 128 + d] = num / den;
    }
}

extern "C" void kernel_launch(void* const* d_in, const int* in_sizes, int n_in,
                              void* d_out, int out_size, void* d_ws, size_t ws_size,
                              hipStream_t stream) {
    const float* query  = (const float*)d_in[0];
    // d_in[1] q_act_seqs: unused (decode: 1 query token per batch)
    const float* ori_kv = (const float*)d_in[2];
    const float* cmp_kv = (const float*)d_in[3];
    const int* ori_bt   = (const int*)d_in[4];
    const int* cmp_bt   = (const int*)d_in[5];
    const int* sink     = (const int*)d_in[6];
    const int* seq      = (const int*)d_in[7];
    const int* cidx     = (const int*)d_in[8];
    const float* scale  = (const float*)d_in[9];
    const int* win      = (const int*)d_in[10];
    const int* ratio    = (const int*)d_in[11];

    const int B    = in_sizes[6];
    const int D    = 128;                 // fixed by cache layout
    const int H    = in_sizes[0] / (B * D);
    const int Hkv  = 4;                   // cache layout [NB,128,2,Hkv,128]
    const int maxblk  = in_sizes[4] / B;
    const int maxcblk = in_sizes[5] / B;
    const int nsel    = in_sizes[8] / B;

    int nsplit = 4;                       // flash-decoding L-partitions
    if ((size_t)B * Hkv * nsplit * 1040 * sizeof(float) > ws_size) nsplit = 1;

    dim3 g1(B * Hkv, nsplit);
    sfa_phase1<<<g1, 256, 0, stream>>>(
        query, ori_kv, cmp_kv, ori_bt, cmp_bt, sink, seq, cidx,
        scale, win, ratio, (float*)d_ws,
        H, Hkv, maxblk, maxcblk, nsel, nsplit);

    sfa_phase2<<<B * Hkv, 256, 0, stream>>>(
        (const float*)d_ws, (float*)d_out, H, Hkv, nsplit);
}